// RoPE_35424890257840
// MI455X (gfx1250) — compile-verified
//
#include <hip/hip_runtime.h>
#include <hip/hip_bf16.h>
#include <cstddef>

// ---------------------------------------------------------------------------
// RoPE-like key/query rotation. Bandwidth-bound streaming kernel.
//   traffic ~540 MB  => ~23us floor @ 23.3 TB/s HBM.
// Strategy: wave32, 16 lanes/token, each lane holds float4 of each 64-block
// of K and Q (8x b128 NT loads, 8x b128 NT stores). locations+freqs staged
// into LDS via CDNA5 async global->LDS loads when available.
// ---------------------------------------------------------------------------

typedef float v4f __attribute__((ext_vector_type(4)));

#define D_DIM     256
#define F_N       18
#define P_N       126          // 7 * 18
#define TOK_BLK   16           // tokens per 256-thread block
#define LOC_ELEMS (TOK_BLK * 7)

#if defined(__gfx1250__) && __has_builtin(__builtin_amdgcn_global_load_async_to_lds_b32) && __has_builtin(__builtin_amdgcn_s_wait_asynccnt)
#define HAVE_ASYNC_LDS 1
#else
#define HAVE_ASYNC_LDS 0
#endif

__device__ __forceinline__ v4f ntload(const v4f* p) {
    return __builtin_nontemporal_load(p);
}
__device__ __forceinline__ void ntstore(v4f* p, v4f v) {
    __builtin_nontemporal_store(v, p);
}

__global__ __launch_bounds__(256) void rope_kernel(
    const float* __restrict__ K, const float* __restrict__ Q,
    const float* __restrict__ loc, const float* __restrict__ freqs,
    float* __restrict__ outK, float* __restrict__ outQ, int nTok)
{
    __shared__ float s_loc[TOK_BLK * 8];   // padded stride 8
    __shared__ float s_frq[F_N];

    const int tid     = threadIdx.x;
    const int tokBase = blockIdx.x * TOK_BLK;

    // ---- Stage loc tile + freqs into LDS --------------------------------
#if HAVE_ASYNC_LDS
    {
        typedef __attribute__((address_space(1))) int gint_t;
        typedef __attribute__((address_space(3))) int lint_t;
        if (tid < LOC_ELEMS) {
            const int row = tid / 7, col = tid % 7;
            if (tokBase + row < nTok) {
                const float* g = loc + (size_t)(tokBase + row) * 7 + col;
                __builtin_amdgcn_global_load_async_to_lds_b32(
                    (gint_t*)g, (lint_t*)&s_loc[row * 8 + col], 0, 0);
            }
        } else if (tid < LOC_ELEMS + F_N) {
            const int j = tid - LOC_ELEMS;
            __builtin_amdgcn_global_load_async_to_lds_b32(
                (gint_t*)(freqs + j), (lint_t*)&s_frq[j], 0, 0);
        }
        __builtin_amdgcn_s_wait_asynccnt(0);
    }
#else
    if (tid < LOC_ELEMS) {
        const int row = tid / 7, col = tid % 7;
        if (tokBase + row < nTok)
            s_loc[row * 8 + col] = loc[(size_t)(tokBase + row) * 7 + col];
    } else if (tid < LOC_ELEMS + F_N) {
        const int j = tid - LOC_ELEMS;
        s_frq[j] = freqs[j];
    }
#endif
    __syncthreads();

    const int tokL = tid >> 4;      // token within block  (0..15)
    const int l    = tid & 15;      // lane within token   (0..15)
    const int tok  = tokBase + tokL;
    if (tok >= nTok) return;

    // ---- stage 1 angles (uniform across the 16 lanes of a token) --------
    const float a4 = s_loc[tokL * 8 + 4];
    const float a5 = s_loc[tokL * 8 + 5];
    const float a6 = s_loc[tokL * 8 + 6];
    float c4, s4, c5, s5, c6, s6;
    __sincosf(a4, &s4, &c4);
    __sincosf(a5, &s5, &c5);
    __sincosf(a6, &s6, &c6);

    // ---- load K/Q: lane owns float4 at local idx 4l of each 64-block ----
    const v4f* kin = (const v4f*)(K + (size_t)tok * D_DIM);
    const v4f* qin = (const v4f*)(Q + (size_t)tok * D_DIM);
    v4f kv[4], qv[4];
#pragma unroll
    for (int b = 0; b < 4; ++b) {
        kv[b] = ntload(kin + b * 16 + l);
        qv[b] = ntload(qin + b * 16 + l);
    }

    // ---- stage 1: triple Givens rotation (lane-local, vector ops) -------
    {
        v4f t;
        // Q: (qx,qy,-a4) -> (qx,qz,-a5) -> (qy,qz,-a6)
        t     = c4 * qv[0] - s4 * qv[1];
        qv[1] = c4 * qv[1] + s4 * qv[0];
        qv[0] = t;
        t     = c5 * qv[0] - s5 * qv[2];
        qv[2] = c5 * qv[2] + s5 * qv[0];
        qv[0] = t;
        t     = c6 * qv[1] - s6 * qv[2];
        qv[2] = c6 * qv[2] + s6 * qv[1];
        qv[1] = t;
        // K: (ky,kz,+a6) -> (kx,kz,+a5) -> (kx,ky,+a4)
        t     = c6 * kv[1] + s6 * kv[2];
        kv[2] = c6 * kv[2] - s6 * kv[1];
        kv[1] = t;
        t     = c5 * kv[0] + s5 * kv[2];
        kv[2] = c5 * kv[2] - s5 * kv[0];
        kv[0] = t;
        t     = c4 * kv[0] + s4 * kv[1];
        kv[1] = c4 * kv[1] - s4 * kv[0];
        kv[0] = t;
    }

    // ---- stage 2 angles: pairs p = b*32 + 2l + k, ang = loc[p/18]*f[p%18]
    float cc[8], ss[8];
#pragma unroll
    for (int b = 0; b < 4; ++b) {
#pragma unroll
        for (int k2 = 0; k2 < 2; ++k2) {
            const int p = b * 32 + 2 * l + k2;
            float cv = 1.0f, sv = 0.0f;   // identity for p >= 126 (tail 252..255)
            if (p < P_N) {
                const float ang = s_loc[tokL * 8 + (p / F_N)] * s_frq[p % F_N];
                __sincosf(ang, &sv, &cv);
            }
            cc[b * 2 + k2] = cv;
            ss[b * 2 + k2] = sv;
        }
    }

    // ---- stage 2: chained pair rotation (matches reference's use of a2) -
#pragma unroll
    for (int b = 0; b < 4; ++b) {
        {
            const float c = cc[2 * b], s = ss[2 * b];
            float ka2 = c * kv[b][0] + s * kv[b][1];
            kv[b][1]  = c * kv[b][1] - s * ka2;
            kv[b][0]  = ka2;
            float qa2 = c * qv[b][0] - s * qv[b][1];
            qv[b][1]  = c * qv[b][1] + s * qa2;
            qv[b][0]  = qa2;
        }
        {
            const float c = cc[2 * b + 1], s = ss[2 * b + 1];
            float ka2 = c * kv[b][2] + s * kv[b][3];
            kv[b][3]  = c * kv[b][3] - s * ka2;
            kv[b][2]  = ka2;
            float qa2 = c * qv[b][2] - s * qv[b][3];
            qv[b][3]  = c * qv[b][3] + s * qa2;
            qv[b][2]  = qa2;
        }
    }

    // ---- NT streaming stores --------------------------------------------
    v4f* kout = (v4f*)(outK + (size_t)tok * D_DIM);
    v4f* qout = (v4f*)(outQ + (size_t)tok * D_DIM);
#pragma unroll
    for (int b = 0; b < 4; ++b) {
        ntstore(kout + b * 16 + l, kv[b]);
        ntstore(qout + b * 16 + l, qv[b]);
    }
}

extern "C" void kernel_launch(void* const* d_in, const int* in_sizes, int n_in,
                              void* d_out, int out_size, void* d_ws, size_t ws_size,
                              hipStream_t stream) {
    const float* K     = (const float*)d_in[0];
    const float* Q     = (const float*)d_in[1];
    const float* loc   = (const float*)d_in[2];
    const float* freqs = (const float*)d_in[3];

    const int nTok = in_sizes[0] / D_DIM;       // 131072
    float* outK = (float*)d_out;                // (K, Q) concatenated flat
    float* outQ = outK + (size_t)nTok * D_DIM;

    const int blocks = (nTok + TOK_BLK - 1) / TOK_BLK;
    hipLaunchKernelGGL(rope_kernel, dim3(blocks), dim3(256), 0, stream,
                       K, Q, loc, freqs, outK, outQ, nTok);
}